// POSSMOutputDecoder_49271864820262
// MI455X (gfx1250) — compile-verified
//
#include <hip/hip_runtime.h>
#include <hip/hip_bf16.h>
#include <math.h>

typedef float v2f __attribute__((ext_vector_type(2)));
typedef float v8f __attribute__((ext_vector_type(8)));

// Problem dims (fixed by reference)
#define B_    8
#define MB_   512
#define IN_   1024
#define EMB_  1024
#define H_    16
#define KVH_  4
#define HD_   64
#define BS_   16
#define KW_   32          // K_HISTORY window
#define NC_   288         // 256 score cols + 32 U cols
#define ROWS_ (B_*MB_)    // 4096

// ---------------------------------------------------------------------------
// 1) qraw = output_queries (16x1024) @ Wq (1024x1024)
// ---------------------------------------------------------------------------
__global__ void qraw_kernel(const float* __restrict__ oq,
                            const float* __restrict__ Wq,
                            float* __restrict__ qraw) {
  int gid = blockIdx.x * blockDim.x + threadIdx.x;   // 16384 total
  int q   = gid >> 10;
  int col = gid & 1023;
  float acc = 0.f;
  for (int i = 0; i < EMB_; ++i)
    acc += oq[q * EMB_ + i] * Wq[i * (H_*HD_) + col];
  qraw[gid] = acc;
}

// ---------------------------------------------------------------------------
// 2) RoPE: qrot[q,h,d] = qraw*cos + rotate_half(qraw)*sin  (rows 0..15 of freqs)
// ---------------------------------------------------------------------------
__global__ void rope_kernel(const float* __restrict__ qraw,
                            const float* __restrict__ fc,
                            const float* __restrict__ fs,
                            float* __restrict__ qrot) {
  int gid = blockIdx.x * blockDim.x + threadIdx.x;   // 16384
  int q = gid >> 10;
  int d = gid & (HD_ - 1);
  float x = qraw[gid];
  float r = (d < HD_/2) ? -qraw[gid + HD_/2] : qraw[gid - HD_/2];
  qrot[gid] = x * fc[q * HD_ + d] + r * fs[q * HD_ + d];
}

// ---------------------------------------------------------------------------
// 3) W2 = Wo (1024x1024) @ Wr (1024x2)
// ---------------------------------------------------------------------------
__global__ void w2_kernel(const float* __restrict__ Wo,
                          const float* __restrict__ Wr,
                          float* __restrict__ W2) {
  int gid = blockIdx.x * blockDim.x + threadIdx.x;   // 2048
  int row = gid >> 1;
  int c   = gid & 1;
  float acc = 0.f;
  for (int e = 0; e < EMB_; ++e)
    acc += Wo[row * EMB_ + e] * Wr[e * 2 + c];
  W2[gid] = acc;
}

// ---------------------------------------------------------------------------
// 4) Folded weights, stored TRANSPOSED: WcombT (288 x 1024), WcombT[col][in]:
//    cols 0..255 : Wkq[in, h*16+q] = sum_d Wk[in,(h/4)*64+d] * qrot[q,h,d]
//    cols 256..287: Wvh[in, h*2+c] = sum_d Wv[in,(h/4)*64+d] * W2[h*64+d, c]
//    Thread map: in = fast index -> coalesced stores.
// ---------------------------------------------------------------------------
__global__ void wcomb_kernel(const float* __restrict__ Wk,
                             const float* __restrict__ Wv,
                             const float* __restrict__ qrot,
                             const float* __restrict__ W2,
                             float* __restrict__ WcombT) {
  int gid = blockIdx.x * blockDim.x + threadIdx.x;   // 288*1024
  int in  = gid & (IN_ - 1);
  int col = gid >> 10;
  float acc = 0.f;
  if (col < 256) {
    int hh = col >> 4, qq = col & 15;
    const float* wk = Wk + in * (KVH_*HD_) + (hh >> 2) * HD_;
    const float* qr = qrot + (qq * H_ + hh) * HD_;
    for (int d = 0; d < HD_; ++d) acc += wk[d] * qr[d];
  } else {
    int j  = col - 256;
    int hh = j >> 1, c = j & 1;
    const float* wv = Wv + in * (KVH_*HD_) + (hh >> 2) * HD_;
    for (int d = 0; d < HD_; ++d) acc += wv[d] * W2[(hh * HD_ + d) * 2 + c];
  }
  WcombT[col * IN_ + in] = acc;
}

// ---------------------------------------------------------------------------
// 5) Main GEMM via V_WMMA_F32_16X16X4_F32:
//    S (4096 x 288) = h_flat (4096 x 1024) @ WcombT^T (1024 x 288)
//    One wave per block; each wave owns a 64x32 output patch (4 M-tiles x 2
//    N-tiles of 16x16), looping K in steps of 4.
//    A layout (16x4 f32): lanes 0-15 hold K=0(.x)/1(.y), lanes 16-31 K=2/3.
//    B layout (4x16):     VGPR0 = rows K=0 (lanes 0-15) / K=2 (16-31),
//                         VGPR1 = rows K=1 / K=3, col = lane%16.
//    With B transposed in memory, each lane's fragment is 2 contiguous floats
//    -> single global_load_b64 per tile per k-step (as for A).
//    C/D (16x16 f32): v8f; lanes 0-15 rows 0-7, lanes 16-31 rows 8-15.
// ---------------------------------------------------------------------------
__global__ void __launch_bounds__(32)
gemm_wmma_kernel(const float* __restrict__ A,      // 4096 x 1024
                 const float* __restrict__ BT,     // 288 x 1024 (col-major B)
                 float* __restrict__ S) {          // 4096 x 288
  const int lane = threadIdx.x;
  const int lrow = lane & 15;
  const int lhi  = lane >> 4;                      // 0 or 1
  const int m0   = blockIdx.x * 64;                // 4 tiles of 16 rows
  const int n0   = blockIdx.y * 32;                // 2 tiles of 16 cols

  v8f acc[4][2];
  const v8f z8 = {0.f,0.f,0.f,0.f,0.f,0.f,0.f,0.f};
#pragma unroll
  for (int i = 0; i < 4; ++i)
#pragma unroll
    for (int j = 0; j < 2; ++j) acc[i][j] = z8;

  const float* arow[4];
#pragma unroll
  for (int i = 0; i < 4; ++i) arow[i] = A + (size_t)(m0 + i * 16 + lrow) * IN_;
  const float* bcol[2];
#pragma unroll
  for (int j = 0; j < 2; ++j) bcol[j] = BT + (size_t)(n0 + j * 16 + lrow) * IN_;

  for (int k0 = 0; k0 < IN_; k0 += 4) {
    const int kk = k0 + (lhi << 1);                // k offset for this half-wave
    v2f b[2];
#pragma unroll
    for (int j = 0; j < 2; ++j)
      b[j] = *reinterpret_cast<const v2f*>(bcol[j] + kk);
    v2f a[4];
#pragma unroll
    for (int i = 0; i < 4; ++i)
      a[i] = *reinterpret_cast<const v2f*>(arow[i] + kk);
#pragma unroll
    for (int i = 0; i < 4; ++i)
#pragma unroll
      for (int j = 0; j < 2; ++j)
        acc[i][j] = __builtin_amdgcn_wmma_f32_16x16x4_f32(
            false, a[i], false, b[j], (short)0, acc[i][j], false, false);
  }

#pragma unroll
  for (int i = 0; i < 4; ++i) {
    const int rowbase = m0 + i * 16 + lhi * 8;
#pragma unroll
    for (int j = 0; j < 2; ++j) {
      const int col = n0 + j * 16 + lrow;
#pragma unroll
      for (int e = 0; e < 8; ++e)
        S[(rowbase + e) * NC_ + col] = acc[i][j][e];
    }
  }
}

// ---------------------------------------------------------------------------
// 6) Windowed softmax-attention + head reduction.
//    Block = 256 threads = (h, q). For each (b,m): over k=0..31, t=m-31+k,
//    s = S[b*512+t, h*16+q]/8 (0 if t<0), u_c = S[..., 256+h*2+c] (0 if t<0).
//    vel[b, m*16+q, c] = sum_h softmax_k(s) . u_c  + br[c]
// ---------------------------------------------------------------------------
__global__ void __launch_bounds__(256)
attn_kernel(const float* __restrict__ S,
            const float* __restrict__ br,
            float* __restrict__ out) {
  __shared__ float lds[H_ * BS_ * 2];
  const int bm = blockIdx.x;           // 0..4095
  const int b  = bm >> 9;
  const int m  = bm & 511;
  const int tid = threadIdx.x;
  const int h  = tid >> 4;
  const int q  = tid & 15;

  float sarr[KW_];
  float smax = 0.f;                    // zero-score pad entries participate
#pragma unroll
  for (int k = 0; k < KW_; ++k) {
    const int t = m - (KW_ - 1) + k;
    float s = 0.f;
    if (t >= 0) s = S[(b * MB_ + t) * NC_ + h * 16 + q] * 0.125f;
    sarr[k] = s;
    smax = fmaxf(smax, s);
  }
  float denom = 0.f, num0 = 0.f, num1 = 0.f;
#pragma unroll
  for (int k = 0; k < KW_; ++k) {
    const int t = m - (KW_ - 1) + k;
    const float e = __expf(sarr[k] - smax);
    denom += e;
    if (t >= 0) {
      const float* u = S + (b * MB_ + t) * NC_ + 256 + h * 2;
      num0 += e * u[0];
      num1 += e * u[1];
    }
  }
  const float inv = 1.f / denom;
  lds[tid * 2 + 0] = num0 * inv;
  lds[tid * 2 + 1] = num1 * inv;
  __syncthreads();

  if (tid < 32) {
    const int qq = tid >> 1, c = tid & 1;
    float acc = 0.f;
#pragma unroll
    for (int hh = 0; hh < H_; ++hh) acc += lds[(hh * 16 + qq) * 2 + c];
    out[((size_t)b * (MB_ * BS_) + m * BS_ + qq) * 2 + c] = acc + br[c];
  }
}

// ---------------------------------------------------------------------------
extern "C" void kernel_launch(void* const* d_in, const int* in_sizes, int n_in,
                              void* d_out, int out_size, void* d_ws, size_t ws_size,
                              hipStream_t stream) {
  const float* h   = (const float*)d_in[0];
  const float* fc  = (const float*)d_in[1];
  const float* fs  = (const float*)d_in[2];
  const float* Wq  = (const float*)d_in[3];
  const float* Wk  = (const float*)d_in[4];
  const float* Wv  = (const float*)d_in[5];
  const float* Wo  = (const float*)d_in[6];
  const float* oq  = (const float*)d_in[7];
  const float* Wr  = (const float*)d_in[8];
  const float* br  = (const float*)d_in[9];
  float* out = (float*)d_out;

  // Workspace layout (floats)
  float* ws     = (float*)d_ws;
  float* qraw   = ws;                        // 16384
  float* qrot   = qraw + 16384;              // 16384
  float* W2     = qrot + 16384;              // 2048
  float* WcombT = W2 + 2048;                 // 288*1024 = 294912
  float* S      = WcombT + 294912;           // 4096*288 = 1179648  (~6 MB total)

  qraw_kernel <<<64,   256, 0, stream>>>(oq, Wq, qraw);
  rope_kernel <<<64,   256, 0, stream>>>(qraw, fc, fs, qrot);
  w2_kernel   <<<8,    256, 0, stream>>>(Wo, Wr, W2);
  wcomb_kernel<<<1152, 256, 0, stream>>>(Wk, Wv, qrot, W2, WcombT);
  gemm_wmma_kernel<<<dim3(ROWS_/64, NC_/32), 32, 0, stream>>>(h, WcombT, S);
  attn_kernel <<<ROWS_, 256, 0, stream>>>(S, br, out);
}